// KINET_DSMC_46600395162347
// MI455X (gfx1250) — compile-verified
//
#include <hip/hip_runtime.h>
#include <math.h>

// Problem constants (from setup_inputs: x/v/a (4,256,1024) f32, rand_u (4,256,8128), dims=arange(128))
#define BS 4
#define CH 256
#define XD 1024
#define ND 128
#define NP 8128
#define TWO_PI_F 6.28318530717958647692f

typedef float v2f __attribute__((ext_vector_type(2)));
typedef float v8f __attribute__((ext_vector_type(8)));

__device__ __forceinline__ v8f wmma_f32_k4(v2f a, v2f b, v8f c) {
  // D = A(16x4 f32) x B(4x16 f32) + C(16x16 f32), wave32, full-precision f32 WMMA
  return __builtin_amdgcn_wmma_f32_16x16x4_f32(false, a, false, b, (short)0, c, false, false);
}

// ---- K0: build inverse map col -> dim-slot (or -1) --------------------------
__global__ void kinet_k0_invmap(const int* __restrict__ dims, int* __restrict__ inv) {
  int t = threadIdx.x;                 // single block of 1024
  inv[t] = -1;
  __syncthreads();
  if (t < ND) inv[dims[t]] = t;
}

// ---- K1: per-batch Gram matrices Gx = Xd^T Xd, Gv = Ad^T Ad via WMMA -------
// grid (64 tiles, 2 mats, 4 batches), block 32 (one wave per 16x16 output tile)
__global__ void kinet_k1_gram(const float* __restrict__ x, const float* __restrict__ a,
                              const int* __restrict__ dims,
                              float* __restrict__ Gx, float* __restrict__ Gv) {
  const int tile = blockIdx.x, mat = blockIdx.y, b = blockIdx.z;
  const int L = threadIdx.x;
  const int i0 = (tile >> 3) * 16, j0 = (tile & 7) * 16;
  const float* src = (mat == 0 ? x : a) + (size_t)b * CH * XD;
  float* G = (mat == 0 ? Gx : Gv) + (size_t)b * ND * ND;

  const int m = L & 15, hi = L >> 4, ksel = hi * 2;
  const int coli = dims[i0 + m];       // fixed column per lane (M index)
  const int colj = dims[j0 + m];

  v8f acc = {0.f, 0.f, 0.f, 0.f, 0.f, 0.f, 0.f, 0.f};
  for (int s = 0; s < CH / 4; ++s) {   // K = 256 channels, 4 per WMMA
    const int c0 = s * 4 + ksel;
    v2f af, bf;
    af[0] = src[(size_t)(c0 + 0) * XD + coli];
    af[1] = src[(size_t)(c0 + 1) * XD + coli];
    bf[0] = src[(size_t)(c0 + 0) * XD + colj];
    bf[1] = src[(size_t)(c0 + 1) * XD + colj];
    acc = wmma_f32_k4(af, bf, acc);
  }
  for (int r = 0; r < 8; ++r)          // C/D layout: VGPR r -> M = r + 8*hi
    G[(size_t)(i0 + hi * 8 + r) * ND + (j0 + m)] = acc[r];
}

// ---- K2: per-batch max of v_r over 128x128 ---------------------------------
__global__ void kinet_k2_vrmax(const float* __restrict__ Gv, float* __restrict__ vrmax) {
  const int b = blockIdx.x, t = threadIdx.x;  // 256 threads
  const float* G = Gv + (size_t)b * ND * ND;
  __shared__ float red[256];
  float mx = 0.f;
  for (int idx = t; idx < ND * ND; idx += 256) {
    const int i = idx >> 7, j = idx & 127;
    const float sq = G[i * ND + i] + G[j * ND + j] - 2.f * G[idx];
    mx = fmaxf(mx, sq > 0.f ? sqrtf(sq) : 0.f);
  }
  red[t] = mx;
  __syncthreads();
  for (int off = 128; off > 0; off >>= 1) {
    if (t < off) red[t] = fmaxf(red[t], red[t + off]);
    __syncthreads();
  }
  if (t == 0) vrmax[b] = red[0];
}

// ---- K3: collision mask, column counts, deterministic compacted pair lists -
// grid (128 cols, 4 batches), block 128 (thread = row i)
__global__ void kinet_k3_mask(const float* __restrict__ Gx, const float* __restrict__ Gv,
                              const float* __restrict__ vrmax,
                              float* __restrict__ CMf, float* __restrict__ Scol,
                              int* __restrict__ cnt, int* __restrict__ plist,
                              float* __restrict__ wlist) {
  const int j = blockIdx.x, b = blockIdx.y, i = threadIdx.x;
  const float* Gxp = Gx + (size_t)b * ND * ND;
  const float* Gvp = Gv + (size_t)b * ND * ND;

  const float sqx = Gxp[i * ND + i] + Gxp[j * ND + j] - 2.f * Gxp[i * ND + j];
  const float xr = sqx > 0.f ? sqrtf(sqx) : 0.f;
  const float sqv = Gvp[i * ND + i] + Gvp[j * ND + j] - 2.f * Gvp[i * ND + j];
  const float vr = sqv > 0.f ? sqrtf(sqv) : 0.f;
  const float mval = vr / vrmax[b] * __expf(-xr);
  const int flag = (i != j) && (mval > 0.5f);

  CMf[(size_t)b * ND * ND + (size_t)i * ND + j] = flag ? 1.f : 0.f;

  __shared__ int sf[ND];
  __shared__ int sp[ND];
  __shared__ float sw[ND];
  const int lo = i < j ? i : j, hj = i < j ? j : i;
  sf[i] = flag;
  sp[i] = lo * ND - (lo * (lo + 1)) / 2 + (hj - lo - 1);      // triu(k=1) pair index
  sw[i] = (i < j ? 1.f : -1.f) * vr * TWO_PI_F;               // signed weight, 2*pi folded in
  __syncthreads();
  if (i == 0) {                                               // serial -> deterministic order
    int cc = 0;
    const int base = (b * ND + j) * (ND - 1);
    for (int k = 0; k < ND; ++k)
      if (sf[k]) { plist[base + cc] = sp[k]; wlist[base + cc] = sw[k]; ++cc; }
    cnt[b * ND + j] = cc;
    Scol[b * ND + j] = (float)cc;
  }
}

// ---- K4: T1 = VV*CM, T3 = XX*CM via WMMA (shared B fragments) --------------
// grid (8 N-tiles, 16 M-tiles, 4 batches), block 32
__global__ void kinet_k4_gemm(const float* __restrict__ x, const float* __restrict__ a,
                              const int* __restrict__ dims, const float* __restrict__ CMf,
                              float* __restrict__ T1, float* __restrict__ T3) {
  const int tj = blockIdx.x, tm = blockIdx.y, b = blockIdx.z;
  const int L = threadIdx.x;
  __shared__ int sd[ND];
  for (int t = L; t < ND; t += 32) sd[t] = dims[t];
  __syncthreads();

  const int m = L & 15, hi = L >> 4, ksel = hi * 2;
  const int m0 = tm * 16, j0 = tj * 16;
  const float* rowV = a + (size_t)b * CH * XD + (size_t)(m0 + m) * XD;
  const float* rowX = x + (size_t)b * CH * XD + (size_t)(m0 + m) * XD;
  const float* CMp = CMf + (size_t)b * ND * ND;

  v8f accV = {0.f, 0.f, 0.f, 0.f, 0.f, 0.f, 0.f, 0.f};
  v8f accX = {0.f, 0.f, 0.f, 0.f, 0.f, 0.f, 0.f, 0.f};
  for (int s = 0; s < ND / 4; ++s) {     // K = 128 collision dims
    const int k0 = s * 4 + ksel;
    v2f bf, av, ax;
    bf[0] = CMp[(size_t)(k0 + 0) * ND + j0 + m];
    bf[1] = CMp[(size_t)(k0 + 1) * ND + j0 + m];
    av[0] = rowV[sd[k0 + 0]];
    av[1] = rowV[sd[k0 + 1]];
    ax[0] = rowX[sd[k0 + 0]];
    ax[1] = rowX[sd[k0 + 1]];
    accV = wmma_f32_k4(av, bf, accV);    // B fragment reused by both products
    accX = wmma_f32_k4(ax, bf, accX);
  }
  for (int r = 0; r < 8; ++r) {
    const size_t o = (size_t)b * CH * ND + (size_t)(m0 + hi * 8 + r) * ND + j0 + m;
    T1[o] = accV[r];
    T3[o] = accX[r];
  }
}

// ---- K5: t2[b,c,j] from compacted lists; skips rand_u entirely if column empty
// grid (128 cols, 4 batches), block 256 (thread = channel)
__global__ void kinet_k5_t2(const float* __restrict__ rand_u, const int* __restrict__ cnt,
                            const int* __restrict__ plist, const float* __restrict__ wlist,
                            float* __restrict__ T2) {
  const int j = blockIdx.x, b = blockIdx.y, c = threadIdx.x;
  const int n = cnt[b * ND + j];
  __shared__ int sp[ND - 1];
  __shared__ float sw[ND - 1];
  const int base = (b * ND + j) * (ND - 1);
  for (int k = c; k < n; k += 256) { sp[k] = plist[base + k]; sw[k] = wlist[base + k]; }
  __syncthreads();
  const float* rr = rand_u + ((size_t)b * CH + c) * NP;
  float acc = 0.f;
  for (int k = 0; k < n; ++k) acc += sw[k] * rr[sp[k]];      // fixed order -> deterministic
  T2[(size_t)b * CH * ND + (size_t)c * ND + j] = acc;
}

// ---- K6: fused final update + scatter into 1024-wide output ----------------
__global__ void kinet_k6_final(const float* __restrict__ x, const float* __restrict__ a,
                               const int* __restrict__ inv, const float* __restrict__ Scol,
                               const float* __restrict__ T1, const float* __restrict__ T2,
                               const float* __restrict__ T3, float* __restrict__ out) {
  const size_t gid = (size_t)blockIdx.x * 256 + threadIdx.x;
  const int col = (int)(gid & 1023);
  const int c = (int)((gid >> 10) & 255);
  const int b = (int)(gid >> 18);
  const float xv = x[gid], av = a[gid];
  const int jn = inv[col];
  float res;
  if (jn < 0) {
    res = xv + av;                                           // untouched columns: x + a
  } else {
    const size_t o = (size_t)b * CH * ND + (size_t)c * ND + jn;
    const float S = Scol[b * ND + jn];
    const float t1 = T1[o], t2 = T2[o], t3 = T3[o];
    const float v_new = av + 0.5f * (S * av - t1) + t2;      // vv_j + Σ delta_v*cmask
    res = (t3 + xv) / (2.f * (S + 1.f)) + 0.5f * xv + 0.5f * v_new;
  }
  out[gid] = res;
}

extern "C" void kernel_launch(void* const* d_in, const int* in_sizes, int n_in,
                              void* d_out, int out_size, void* d_ws, size_t ws_size,
                              hipStream_t stream) {
  const float* x      = (const float*)d_in[0];
  // d_in[1] (v) is unused by the reference
  const float* a      = (const float*)d_in[2];
  const float* rand_u = (const float*)d_in[3];
  const int*   dims   = (const int*)d_in[4];
  float* out = (float*)d_out;
  (void)in_sizes; (void)n_in; (void)out_size; (void)ws_size;

  // Workspace carve-up (all 256B-aligned sizes)
  char* p = (char*)d_ws;
  float* Gx    = (float*)p; p += (size_t)BS * ND * ND * 4;        // 256 KB
  float* Gv    = (float*)p; p += (size_t)BS * ND * ND * 4;        // 256 KB
  float* CMf   = (float*)p; p += (size_t)BS * ND * ND * 4;        // 256 KB
  float* T1    = (float*)p; p += (size_t)BS * CH * ND * 4;        // 512 KB
  float* T2    = (float*)p; p += (size_t)BS * CH * ND * 4;        // 512 KB
  float* T3    = (float*)p; p += (size_t)BS * CH * ND * 4;        // 512 KB
  int*   plist = (int*)p;   p += (size_t)BS * ND * (ND - 1) * 4;  // 254 KB
  float* wlist = (float*)p; p += (size_t)BS * ND * (ND - 1) * 4;  // 254 KB
  float* Scol  = (float*)p; p += (size_t)BS * ND * 4;
  int*   cnt   = (int*)p;   p += (size_t)BS * ND * 4;
  int*   inv   = (int*)p;   p += (size_t)XD * 4;
  float* vrmax = (float*)p; p += 256;

  kinet_k0_invmap<<<1, 1024, 0, stream>>>(dims, inv);
  kinet_k1_gram<<<dim3(64, 2, BS), 32, 0, stream>>>(x, a, dims, Gx, Gv);
  kinet_k2_vrmax<<<BS, 256, 0, stream>>>(Gv, vrmax);
  kinet_k3_mask<<<dim3(ND, BS), ND, 0, stream>>>(Gx, Gv, vrmax, CMf, Scol, cnt, plist, wlist);
  kinet_k4_gemm<<<dim3(ND / 16, CH / 16, BS), 32, 0, stream>>>(x, a, dims, CMf, T1, T3);
  kinet_k5_t2<<<dim3(ND, BS), 256, 0, stream>>>(rand_u, cnt, plist, wlist, T2);
  kinet_k6_final<<<(BS * CH * XD) / 256, 256, 0, stream>>>(x, a, inv, Scol, T1, T2, T3, out);
}